// PointPillar_73985106641253
// MI455X (gfx1250) — compile-verified
//
#include <hip/hip_runtime.h>

// ---- problem constants (match reference) ----
#define GX 440
#define GY 500
#define GZ 1
#define NC (GX * GY * GZ)        // 220000 cells
#define MAXV 40000
#define MAXP 32
#define VOXSLOTS (MAXV * MAXP)   // 1,280,000 (voxel, slot) pairs
// d_out layout (floats), concatenated in reference return order
#define VOX_OFF  0
#define COOR_OFF (MAXV * MAXP * 4)       // 5,120,000
#define NPV_OFF  (COOR_OFF + MAXV * 3)   // 5,240,000
#define VN_OFF   (NPV_OFF + MAXV)        // 5,280,000

#define SCAN_B 1024
#define SEL_B  256

// ---- CDNA5 async global->LDS path (guarded; falls back if not declared) ----
#if defined(__AMDGCN__) && __has_builtin(__builtin_amdgcn_global_load_async_to_lds_b32)
#define HAVE_ASYNC_LDS 1
typedef __attribute__((address_space(1))) int g1int;
typedef __attribute__((address_space(3))) int l3int;
#else
#define HAVE_ASYNC_LDS 0
#endif

__device__ __forceinline__ void async_wait0() {
#if __has_builtin(__builtin_amdgcn_s_wait_asynccnt)
    __builtin_amdgcn_s_wait_asynccnt(0);
#elif defined(__AMDGCN__)
    asm volatile("s_wait_asynccnt 0x0" ::: "memory");
#endif
}

// ---------------- init kernels ----------------
__global__ void k_zero_f(float* p, int n) {
    int i = blockIdx.x * blockDim.x + threadIdx.x;
    if (i < n) p[i] = 0.0f;
}

__global__ void k_fill_i(int* p, int n, int v) {
    int i = blockIdx.x * blockDim.x + threadIdx.x;
    if (i < n) p[i] = v;
}

// init per-cell state; also zeroes the creator bitmask (nw <= NC)
__global__ void k_init_cells(int* first, int* count, int* cursor, int* voxid,
                             unsigned* fbits, int nw, int n_points) {
    int c = blockIdx.x * blockDim.x + threadIdx.x;
    if (c < NC) {
        first[c]  = n_points;  // +inf sentinel (all real indices < N)
        count[c]  = 0;
        cursor[c] = 0;
        voxid[c]  = -1;
    }
    if (c < nw) fbits[c] = 0u;
}

// ---------------- pass A: bin points (deterministic atomics) ----------------
__global__ void k_points_a(const float4* __restrict__ pts, int N,
                           int* __restrict__ lin_out,
                           int* __restrict__ first, int* __restrict__ count) {
    int i = blockIdx.x * blockDim.x + threadIdx.x;
    if (i >= N) return;
    if (i + 4096 < N) __builtin_prefetch(pts + i + 4096, 0, 0); // global_prefetch_b8
    float4 p = pts[i];
    int ix = (int)floorf((p.x - 0.0f)  / 0.16f);
    int iy = (int)floorf((p.y + 40.0f) / 0.16f);
    int iz = (int)floorf((p.z + 3.0f)  / 4.0f);
    bool valid = (ix >= 0) & (ix < GX) & (iy >= 0) & (iy < GY) &
                 (iz >= 0) & (iz < GZ);
    int lin = ix * (GY * GZ) + iy * GZ + iz;
    lin_out[i] = valid ? lin : -1;
    if (valid) {
        atomicMin(&first[lin], i);   // commutative -> deterministic
        atomicAdd(&count[lin], 1);   // commutative -> deterministic
    }
}

// ---------------- creator bit at position first[cell] (packed) ----------------
__global__ void k_flags(const int* __restrict__ first,
                        const int* __restrict__ count,
                        unsigned* __restrict__ fbits) {
    int c = blockIdx.x * blockDim.x + threadIdx.x;
    if (c < NC && count[c] > 0) {
        int f = first[c];
        atomicOr(&fbits[f >> 5], 1u << (f & 31));
    }
}

// ---------------- LDS exclusive scans (wave32 WGP) ----------------
// stage one word per thread into LDS (async global->LDS when available);
// builtin is width-typed: b32 takes int* (as1 src, as3 dst)
#define STAGE_TO_LDS(in, gid, n, sh)                                           \
    do {                                                                       \
        if ((gid) < (n)) {                                                     \
            __builtin_amdgcn_global_load_async_to_lds_b32(                     \
                (g1int*)(unsigned long long)((in) + (gid)),                    \
                (l3int*)(unsigned)(unsigned long long)&(sh)[threadIdx.x],      \
                0, 0);                                                         \
        } else {                                                               \
            (sh)[threadIdx.x] = 0;                                             \
        }                                                                      \
        async_wait0();                                                         \
        __syncthreads();                                                       \
    } while (0)

// scan of popcounts of a bitmask array (rank base per 32-bit word)
__global__ void k_scan_popc(const unsigned* __restrict__ in, int* __restrict__ out,
                            int n, int* __restrict__ bsums) {
    __shared__ int sh[SCAN_B];
    int gid = blockIdx.x * SCAN_B + threadIdx.x;
#if HAVE_ASYNC_LDS
    STAGE_TO_LDS(in, gid, n, sh);
    int x = __popc((unsigned)sh[threadIdx.x]);
#else
    int x = (gid < n) ? __popc(in[gid]) : 0;
#endif
    int val = x;
    sh[threadIdx.x] = x;
    __syncthreads();
    for (int off = 1; off < SCAN_B; off <<= 1) {
        int y = (threadIdx.x >= off) ? sh[threadIdx.x - off] : 0;
        __syncthreads();
        val += y;
        sh[threadIdx.x] = val;
        __syncthreads();
    }
    if (gid < n) out[gid] = val - x;                       // exclusive in block
    if (threadIdx.x == SCAN_B - 1) bsums[blockIdx.x] = val;
}

// plain int scan (for per-cell counts -> segment offsets)
__global__ void k_scan_block(const int* __restrict__ in, int* __restrict__ out,
                             int n, int* __restrict__ bsums) {
    __shared__ int sh[SCAN_B];
    int gid = blockIdx.x * SCAN_B + threadIdx.x;
#if HAVE_ASYNC_LDS
    STAGE_TO_LDS(in, gid, n, sh);
    int x = sh[threadIdx.x];
#else
    int x = (gid < n) ? in[gid] : 0;
    sh[threadIdx.x] = x;
    __syncthreads();
#endif
    int val = x;
    sh[threadIdx.x] = x;
    __syncthreads();
    for (int off = 1; off < SCAN_B; off <<= 1) {
        int y = (threadIdx.x >= off) ? sh[threadIdx.x - off] : 0;
        __syncthreads();
        val += y;
        sh[threadIdx.x] = val;
        __syncthreads();
    }
    if (gid < n) out[gid] = val - x;
    if (threadIdx.x == SCAN_B - 1) bsums[blockIdx.x] = val;
}

__global__ void k_scan_bsums(int* __restrict__ bs, int nb, int* __restrict__ total) {
    __shared__ int sh[SCAN_B];
    __shared__ int carry_s;
    if (threadIdx.x == 0) carry_s = 0;
    __syncthreads();
    for (int base = 0; base < nb; base += SCAN_B) {
        int gid = base + threadIdx.x;
        int x = (gid < nb) ? bs[gid] : 0;
        int val = x;
        sh[threadIdx.x] = x;
        __syncthreads();
        for (int off = 1; off < SCAN_B; off <<= 1) {
            int y = (threadIdx.x >= off) ? sh[threadIdx.x - off] : 0;
            __syncthreads();
            val += y;
            sh[threadIdx.x] = val;
            __syncthreads();
        }
        int c = carry_s;
        if (gid < nb) bs[gid] = c + val - x;
        __syncthreads();
        if (threadIdx.x == SCAN_B - 1) carry_s = c + val;
        __syncthreads();
    }
    if (threadIdx.x == 0 && total) *total = carry_s;
}

// ---------------- per-cell: FCFS rank -> voxel id, coors, npv ----------------
__global__ void k_cells(const int* __restrict__ first,
                        const int* __restrict__ count,
                        const unsigned* __restrict__ fbits,
                        const int* __restrict__ Sw,     // block-excl popc scan
                        const int* __restrict__ bsumsW, // scanned block sums
                        int* __restrict__ voxid,
                        float* __restrict__ out) {
    int c = blockIdx.x * blockDim.x + threadIdx.x;
    if (c >= NC) return;
    int n = count[c];
    if (n <= 0) return;
    int f = first[c];
    int wi = f >> 5, bi = f & 31;
    int r = Sw[wi] + bsumsW[wi >> 10] +
            __popc(fbits[wi] & ((1u << bi) - 1u));       // FCFS rank
    if (r >= MAXV) return;                               // voxel dropped
    voxid[c] = r;
    int ix = c / (GY * GZ);
    int rem = c % (GY * GZ);
    int iy = rem / GZ;
    int iz = rem % GZ;
    out[COOR_OFF + 3 * r + 0] = (float)ix;
    out[COOR_OFF + 3 * r + 1] = (float)iy;
    out[COOR_OFF + 3 * r + 2] = (float)iz;
    out[NPV_OFF + r] = (float)(n < MAXP ? n : MAXP);
}

__global__ void k_voxnum(const int* __restrict__ total, float* __restrict__ out) {
    int t = total[0];
    out[VN_OFF] = (float)(t < MAXV ? t : MAXV);
}

// ---------------- pass B: scatter point ids into per-cell segments ----------------
__global__ void k_points_b(const int* __restrict__ lin, int N,
                           const int* __restrict__ offs,
                           const int* __restrict__ bsumsC,
                           int* __restrict__ cursor, int* __restrict__ seg) {
    int i = blockIdx.x * blockDim.x + threadIdx.x;
    if (i >= N) return;
    int c = lin[i];
    if (c >= 0) {
        int pos = offs[c] + bsumsC[c >> 10] + atomicAdd(&cursor[c], 1);
        seg[pos] = i;                        // order canonicalized below
    }
}

// ---------------- per-cell: select 32 smallest indices (deterministic slots) ----
// per-thread 32-entry working set lives in LDS, [slot][tid] layout so the 32
// lanes of a wave hit 32 consecutive banks (conflict-free on 64-bank LDS)
__global__ void k_select(const int* __restrict__ count,
                         const int* __restrict__ offs,
                         const int* __restrict__ bsumsC,
                         const int* __restrict__ voxid,
                         const int* __restrict__ seg,
                         int* __restrict__ voxpt) {
    __shared__ int sb[MAXP * SEL_B];            // 32 KB of the 320 KB WGP LDS
    int tid = threadIdx.x;
    int c = blockIdx.x * SEL_B + tid;
    if (c >= NC) return;
    int n = count[c];
    int v = voxid[c];
    if (n <= 0 || v < 0) return;
    int base = offs[c] + bsumsC[c >> 10];
    int m = 0;
    for (int k = 0; k < n; ++k) {
        int id = seg[base + k];
        if (m < MAXP) {
            int j = m++;
            while (j > 0 && sb[(j - 1) * SEL_B + tid] > id) {
                sb[j * SEL_B + tid] = sb[(j - 1) * SEL_B + tid];
                --j;
            }
            sb[j * SEL_B + tid] = id;
        } else if (id < sb[(MAXP - 1) * SEL_B + tid]) {
            int j = MAXP - 1;
            while (j > 0 && sb[(j - 1) * SEL_B + tid] > id) {
                sb[j * SEL_B + tid] = sb[(j - 1) * SEL_B + tid];
                --j;
            }
            sb[j * SEL_B + tid] = id;
        }
    }
    int w = n < MAXP ? n : MAXP;
    for (int s = 0; s < w; ++s) voxpt[v * MAXP + s] = sb[s * SEL_B + tid];
}

// ---------------- final fill: coalesced 128-bit copies ----------------
__global__ void k_fill_vox(const int* __restrict__ voxpt,
                           const float4* __restrict__ pts,
                           float4* __restrict__ out_vox) {
    int t = blockIdx.x * blockDim.x + threadIdx.x;
    if (t >= VOXSLOTS) return;
    int idx = voxpt[t];
    if (idx >= 0) out_vox[t] = pts[idx];   // global_load_b128 / global_store_b128
}

extern "C" void kernel_launch(void* const* d_in, const int* in_sizes, int n_in,
                              void* d_out, int out_size, void* d_ws, size_t ws_size,
                              hipStream_t stream) {
    const float4* pts = (const float4*)d_in[0];
    const int N  = in_sizes[0] / 4;
    const int NW = (N + 31) / 32;          // words in creator bitmask
    float* out = (float*)d_out;

    // workspace carve-up (ints)
    int* w       = (int*)d_ws;
    int* first   = w;  w += NC;
    int* count   = w;  w += NC;
    int* cursor  = w;  w += NC;
    int* voxid   = w;  w += NC;
    int* offs    = w;  w += NC + 64;
    int* bsumsW  = w;  w += 4096;
    int* bsumsC  = w;  w += 4096;
    int* total   = w;  w += 64;
    unsigned* fbits = (unsigned*)w;  w += NW + 64;
    int* Sw      = w;  w += NW + 64;
    int* voxpt   = w;  w += VOXSLOTS;
    int* lin     = w;  w += N;
    int* seg     = w;  w += N;

    const int B = 256;
    int gN   = (N + B - 1) / B;
    int gNC  = (NC + B - 1) / B;
    int gVS  = (VOXSLOTS + B - 1) / B;
    int gOut = (out_size + B - 1) / B;
    int gSel = (NC + SEL_B - 1) / SEL_B;
    int nbW  = (NW + SCAN_B - 1) / SCAN_B;   // 123 for N=4M
    int nbC  = (NC + SCAN_B - 1) / SCAN_B;   // 215

    // init (re-done every call; graph-safe)
    k_zero_f<<<gOut, B, 0, stream>>>(out, out_size);
    k_init_cells<<<gNC, B, 0, stream>>>(first, count, cursor, voxid, fbits, NW, N);
    k_fill_i<<<gVS, B, 0, stream>>>(voxpt, VOXSLOTS, -1);

    // pass A: bin points
    k_points_a<<<gN, B, 0, stream>>>(pts, N, lin, first, count);

    // FCFS voxel ranks: packed creator bits, popcount scan over N/32 words
    k_flags<<<gNC, B, 0, stream>>>(first, count, fbits);
    k_scan_popc<<<nbW, SCAN_B, 0, stream>>>(fbits, Sw, NW, bsumsW);
    k_scan_bsums<<<1, SCAN_B, 0, stream>>>(bsumsW, nbW, total);

    // per-cell metadata + voxel_num (block prefix added inline from bsumsW)
    k_cells<<<gNC, B, 0, stream>>>(first, count, fbits, Sw, bsumsW, voxid, out);
    k_voxnum<<<1, 1, 0, stream>>>(total, out);

    // segment offsets: exclusive scan of per-cell counts (prefix via bsumsC)
    k_scan_block<<<nbC, SCAN_B, 0, stream>>>(count, offs, NC, bsumsC);
    k_scan_bsums<<<1, SCAN_B, 0, stream>>>(bsumsC, nbC, nullptr);

    // pass B: scatter ids, then deterministic top-32 selection per cell (LDS)
    k_points_b<<<gN, B, 0, stream>>>(lin, N, offs, bsumsC, cursor, seg);
    k_select<<<gSel, SEL_B, 0, stream>>>(count, offs, bsumsC, voxid, seg, voxpt);

    // final coalesced fill of voxels
    k_fill_vox<<<gVS, B, 0, stream>>>(voxpt, pts, (float4*)(out + VOX_OFF));
}